// CFConv_27994596835767
// MI455X (gfx1250) — compile-verified
//
#include <hip/hip_runtime.h>

#define NF 64
#define HD 64
#define OD 128

typedef float v2f __attribute__((ext_vector_type(2)));
typedef float v8f __attribute__((ext_vector_type(8)));

__device__ __forceinline__ float sspf(float x) {
    // shifted softplus: log(1+exp(x)) - log(2), numerically stable
    float ax = fabsf(x);
    return fmaxf(x, 0.0f) + __logf(1.0f + __expf(-ax)) - 0.6931471805599453f;
}

__global__ void __launch_bounds__(256) zero_f32(float* __restrict__ p, int n) {
    int i = blockIdx.x * 256 + threadIdx.x;
    if (i < n) p[i] = 0.0f;
}

__global__ void __launch_bounds__(256) cfconv_kernel(
    const float* __restrict__ feat,      // [N, 128]
    const float* __restrict__ pos,       // [N, 3]
    const long long* __restrict__ ei,    // [2, E]
    const float* __restrict__ W1,        // [64, 64]
    const float* __restrict__ b1,        // [64]
    const float* __restrict__ W2,        // [128, 64]
    const float* __restrict__ b2,        // [128]
    const float* __restrict__ lbp, const float* __restrict__ ubp,
    const float* __restrict__ gp,
    float* __restrict__ out,             // [N, 128]
    long long E)
{
    __shared__ float w1s[HD * NF];   // [k][n] = W1[n][k]  (B-operand layout, GEMM1)
    __shared__ float w2s[HD * OD];   // [k][n] = W2[n][k]  (B-operand layout, GEMM2)
    __shared__ float hs[8 * 272];    // per-wave h chunk, transposed [k_local][17]

    const int tid = threadIdx.x;

    // Stage transposed weights into LDS (one-time, coalesced global reads)
    for (int i = tid; i < HD * NF; i += 256) {
        int n = i >> 6, k = i & 63;
        w1s[k * NF + n] = W1[i];
    }
    for (int i = tid; i < OD * HD; i += 256) {
        int n = i >> 6, k = i & 63;
        w2s[k * OD + n] = W2[i];
    }
    __syncthreads();

    const float lb    = lbp[0];
    const float gamma = gp[0];
    const float step  = (ubp[0] - lb) * (1.0f / (float)(NF - 1));

    const int lane = tid & 31;
    const int m    = lane & 15;   // A-operand row / C-operand column-within-tile
    const int hi   = lane >> 4;   // 0: lanes 0-15, 1: lanes 16-31
    const int aoff = hi << 1;     // K offset (+2) for upper half-wave
    float* hsw = &hs[(tid >> 5) * 272];

    // Bias values per lane (column n = nt*16 + m), loop invariant
    float bias1[4], bias2[8];
#pragma unroll
    for (int t = 0; t < 4; ++t) bias1[t] = b1[t * 16 + m];
#pragma unroll
    for (int t = 0; t < 8; ++t) bias2[t] = b2[t * 16 + m];

    const int ntiles = (int)((E + 15) >> 4);
    const int wave   = (int)((blockIdx.x * blockDim.x + tid) >> 5);
    const int nwaves = (int)((gridDim.x * blockDim.x) >> 5);

    for (int tile = wave; tile < ntiles; tile += nwaves) {   // wave-uniform loop
        long long e  = (long long)tile * 16 + m;
        long long ec = (e < E) ? e : 0;
        int src = (int)ei[ec];
        int tgt = (int)ei[E + ec];
        float dx = pos[src * 3 + 0] - pos[tgt * 3 + 0];
        float dy = pos[src * 3 + 1] - pos[tgt * 3 + 1];
        float dz = pos[src * 3 + 2] - pos[tgt * 3 + 2];
        float dist = sqrtf(dx * dx + dy * dy + dz * dz);

        // RBF features directly in WMMA f32 16x16x4 A layout:
        // a1r[2*kb+j] = exp(-gamma*(dist - mu(4*kb + aoff + j))^2)
        float a1r[32];
#pragma unroll
        for (int kb = 0; kb < 16; ++kb) {
#pragma unroll
            for (int j = 0; j < 2; ++j) {
                int k = 4 * kb + aoff + j;
                float t = dist - (lb + (float)k * step);
                a1r[2 * kb + j] = __expf(-gamma * t * t);
            }
        }

        // GEMM2 accumulators (8 N-tiles of 16x16), init with bias b2
        v8f acc2[8];
#pragma unroll
        for (int t = 0; t < 8; ++t)
#pragma unroll
            for (int r = 0; r < 8; ++r) acc2[t][r] = bias2[t];

#pragma unroll
        for (int nt1 = 0; nt1 < 4; ++nt1) {
            // ---- GEMM1 tile nt1: expanded[16x64] x W1^T[:,16nt1:16nt1+16] ----
            v8f acc1;
#pragma unroll
            for (int r = 0; r < 8; ++r) acc1[r] = bias1[nt1];
            int n1 = nt1 * 16 + m;
#pragma unroll
            for (int kb = 0; kb < 16; ++kb) {
                v2f A; A.x = a1r[2 * kb]; A.y = a1r[2 * kb + 1];
                v2f B;
                B.x = w1s[(4 * kb + aoff)     * NF + n1];
                B.y = w1s[(4 * kb + aoff + 1) * NF + n1];
                acc1 = __builtin_amdgcn_wmma_f32_16x16x4_f32(
                    false, A, false, B, (short)0, acc1, false, false);
            }
            // softplus, stage h chunk transposed into per-wave LDS: hs[k_local][m_row]
#pragma unroll
            for (int r = 0; r < 8; ++r) {
                hsw[m * 17 + (r + 8 * hi)] = sspf(acc1[r]);
            }
            // ---- GEMM2 partial: this chunk contributes K = 16*nt1 .. 16*nt1+15 ----
#pragma unroll
            for (int kbl = 0; kbl < 4; ++kbl) {
                v2f A2;
                A2.x = hsw[(4 * kbl + aoff)     * 17 + m];
                A2.y = hsw[(4 * kbl + aoff + 1) * 17 + m];
                int k2 = 16 * nt1 + 4 * kbl;
#pragma unroll
                for (int nt2 = 0; nt2 < 8; ++nt2) {
                    int n2 = nt2 * 16 + m;
                    v2f B2;
                    B2.x = w2s[(k2 + aoff)     * OD + n2];
                    B2.y = w2s[(k2 + aoff + 1) * OD + n2];
                    acc2[nt2] = __builtin_amdgcn_wmma_f32_16x16x4_f32(
                        false, A2, false, B2, (short)0, acc2[nt2], false, false);
                }
            }
        }

        // filt = softplus(acc2) ; gather feat[target], modulate, scatter-add
#pragma unroll
        for (int r = 0; r < 8; ++r) {
            int mm = r + 8 * hi;                 // output row (edge within tile)
            int tg = __shfl(tgt, mm, 32);        // target node of that edge
            if ((long long)tile * 16 + mm < E) {
#pragma unroll
                for (int nt2 = 0; nt2 < 8; ++nt2) {
                    int n2 = nt2 * 16 + m;
                    float f = sspf(acc2[nt2][r]);
                    float msg = feat[(size_t)tg * OD + n2] * f;
                    unsafeAtomicAdd(&out[(size_t)tg * OD + n2], msg);
                }
            }
        }
    }
}

extern "C" void kernel_launch(void* const* d_in, const int* in_sizes, int n_in,
                              void* d_out, int out_size, void* d_ws, size_t ws_size,
                              hipStream_t stream) {
    const float*     feat = (const float*)d_in[0];
    const float*     pos  = (const float*)d_in[1];
    const long long* ei   = (const long long*)d_in[2];
    const float*     W1   = (const float*)d_in[3];
    const float*     b1   = (const float*)d_in[4];
    const float*     W2   = (const float*)d_in[5];
    const float*     b2   = (const float*)d_in[6];
    const float*     lbp  = (const float*)d_in[7];
    const float*     ubp  = (const float*)d_in[8];
    const float*     gp   = (const float*)d_in[9];
    float* out = (float*)d_out;

    long long E = (long long)in_sizes[2] / 2;

    // out is poisoned by the harness; zero it (segment_sum starts from 0)
    int nz = out_size;
    zero_f32<<<(nz + 255) / 256, 256, 0, stream>>>(out, nz);

    int ntiles = (int)((E + 15) >> 4);
    int blocks = (ntiles + 7) / 8;   // 8 waves per block
    if (blocks > 4096) blocks = 4096;
    if (blocks < 1) blocks = 1;
    cfconv_kernel<<<blocks, 256, 0, stream>>>(feat, pos, ei, W1, b1, W2, b2,
                                              lbp, ubp, gp, out, E);
}